// GraphDA_5789615915324
// MI455X (gfx1250) — compile-verified
//
#include <hip/hip_runtime.h>
#include <math.h>

typedef __attribute__((ext_vector_type(2))) float v2f;
typedef __attribute__((ext_vector_type(8))) float v8f;

#define LRELU 0.2f

__device__ __forceinline__ void atomic_max_f32(float* addr, float val) {
  // signed-max for nonneg floats, unsigned-min for negatives; init must be -inf
  if (val >= 0.f) atomicMax((int*)addr, __float_as_int(val));
  else            atomicMin((unsigned int*)addr, __float_as_uint(val));
}

// ---------------------------------------------------------------------------
// WMMA fp32 GEMM: C[M,Nout] = A[M,K] @ B[K,Nout], row-major.
// One workgroup per 16-row M tile; the 16xK A tile is staged into LDS once
// via async global->LDS b128 copies (ASYNCcnt), then each wave computes one
// 16-wide N tile with V_WMMA_F32_16X16X4_F32, reading A from LDS (padded
// rows -> conflict-free ds_load) and B from global (L2-resident weights).
// A 16x4 layout: lane L -> M=L%16, VGPR0 K=(L<16?0:2), VGPR1 K=(L<16?1:3)
// B 4x16 layout: lane L -> N=L%16, VGPR0 K=(L<16?0:2), VGPR1 K=(L<16?1:3)
// C 16x16 layout: VGPR v, lane L -> M = v + (L>=16?8:0), N = L%16
// ---------------------------------------------------------------------------
#define KMAX 512
#define KPAD 4   // dword padding per LDS row (bank-conflict avoidance)

__global__ void __launch_bounds__(256)
gemm_wmma_f32(const float* __restrict__ A, const float* __restrict__ B,
              float* __restrict__ C, int K, int Nout) {
  __shared__ float As[16 * (KMAX + KPAD)];
  const int KP = K + KPAD;
  const int tm = blockIdx.x;
  const int tid = threadIdx.x;
  const int wave = tid >> 5;
  const int lane = tid & 31;

  // ---- async stage of the 16xK A tile into LDS (16-byte chunks) ----
  const int cpr = K >> 2;           // 16B chunks per row
  const int chunks = cpr << 4;      // 16 rows
  for (int t = tid; t < chunks; t += blockDim.x) {
    int row = t / cpr;
    int ch  = t - row * cpr;
    const float* g = A + (size_t)(tm * 16 + row) * K + (ch << 2);
    unsigned lds = (unsigned)(uintptr_t)(As + row * KP + (ch << 2));
    asm volatile("global_load_async_to_lds_b128 %0, %1, off"
                 :: "v"(lds), "v"(g) : "memory");
  }
  asm volatile("s_wait_asynccnt 0" ::: "memory");
  __syncthreads();

  // ---- per-wave 16x16 WMMA tile ----
  const int tn = wave;              // blockDim.x == (Nout/16)*32
  const int r  = lane & 15;
  const int kh = (lane >> 4) << 1;  // 0 or 2
  const float* arow = As + r * KP + kh;                       // LDS
  const float* bcol = B + (size_t)kh * Nout + (tn * 16 + r);  // global
  __builtin_prefetch(bcol, 0, 1);   // global_prefetch_b8
  v8f acc = {};
  for (int k = 0; k < K; k += 4) {
    v2f a, b;
    a.x = arow[k];
    a.y = arow[k + 1];
    b.x = bcol[(size_t)k * Nout];
    b.y = bcol[(size_t)(k + 1) * Nout];
    acc = __builtin_amdgcn_wmma_f32_16x16x4_f32(
        /*neg_a=*/false, a, /*neg_b=*/false, b,
        /*c_mod=*/(short)0, acc, /*reuse_a=*/false, /*reuse_b=*/false);
  }
  const int rowbase = tm * 16 + ((lane >> 4) << 3);
  float* cp = C + (size_t)rowbase * Nout + (tn * 16 + r);
#pragma unroll
  for (int v = 0; v < 8; ++v) cp[(size_t)v * Nout] = acc[v];
}

// per-node attention scores: ssrc[i] = h[i]·a_src, sdst[i] = h[i]·a_dst
__global__ void node_scores(const float* __restrict__ H, const float* __restrict__ as,
                            const float* __restrict__ ad, float* __restrict__ ssrc,
                            float* __restrict__ sdst, int N, int D) {
  int wid  = (blockIdx.x * blockDim.x + threadIdx.x) >> 5;
  int lane = threadIdx.x & 31;
  if (wid >= N) return;
  const float* h = H + (size_t)wid * D;
  float s0 = 0.f, s1 = 0.f;
  for (int d = lane; d < D; d += 32) {
    float v = h[d];
    s0 += v * as[d];
    s1 += v * ad[d];
  }
  for (int o = 16; o; o >>= 1) {
    s0 += __shfl_down(s0, o, 32);
    s1 += __shfl_down(s1, o, 32);
  }
  if (lane == 0) { ssrc[wid] = s0; sdst[wid] = s1; }
}

__global__ void init_ms(float* __restrict__ m, float* __restrict__ s, int N) {
  int i = blockIdx.x * blockDim.x + threadIdx.x;
  if (i < N) { m[i] = -INFINITY; s[i] = 0.f; }
}

__global__ void zero_f32(float* __restrict__ p, size_t n) {
  size_t i = (size_t)blockIdx.x * blockDim.x + threadIdx.x;
  size_t stride = (size_t)gridDim.x * blockDim.x;
  for (; i < n; i += stride) p[i] = 0.f;
}

// pass 1: alpha = leaky_relu(ssrc[src] + sdst[dst] + edge_term); segment max
__global__ void edge_alpha_max(const int* __restrict__ ei, int E, int N,
                               const float* __restrict__ ssrc, const float* __restrict__ sdst,
                               int has_e, const float* __restrict__ ea,
                               const float* __restrict__ mean_p, const float* __restrict__ ce_p,
                               float* __restrict__ alpha, float* __restrict__ mmax) {
  int idx = blockIdx.x * blockDim.x + threadIdx.x;
  if (idx >= E + N) return;
  int s, d; float et = 0.f;
  if (idx < E) {
    s = ei[idx]; d = ei[E + idx];
    if (has_e) et = ea[idx] * ce_p[0];
  } else {
    s = d = idx - E;                  // self loop
    if (has_e) et = mean_p[0] * ce_p[0];
  }
  float a = ssrc[s] + sdst[d] + et;
  a = (a >= 0.f) ? a : LRELU * a;
  alpha[idx] = a;
  atomic_max_f32(&mmax[d], a);
}

// pass 2: e = exp(alpha - m[dst]); segment sum
__global__ void edge_exp(const int* __restrict__ ei, int E, int N,
                         float* __restrict__ alpha, const float* __restrict__ mmax,
                         float* __restrict__ ssum) {
  int idx = blockIdx.x * blockDim.x + threadIdx.x;
  if (idx >= E + N) return;
  int d = (idx < E) ? ei[E + idx] : idx - E;
  float e = expf(alpha[idx] - mmax[d]);
  alpha[idx] = e;
  atomicAdd(&ssum[d], e);
}

// pass 3: agg[dst,:] += h[src,:] * e/(s[dst]+eps); one thread per (edge, 4-dim)
__global__ void edge_agg4(const int* __restrict__ ei, int E, int N, int Dlog2,
                          const float* __restrict__ H, const float* __restrict__ eexp,
                          const float* __restrict__ ssum, float* __restrict__ agg) {
  size_t idx = (size_t)blockIdx.x * blockDim.x + threadIdx.x;
  int q = Dlog2 - 2;                       // log2(D/4)
  size_t total = ((size_t)(E + N)) << q;
  if (idx >= total) return;
  int e  = (int)(idx >> q);
  int c4 = ((int)idx & ((1 << q) - 1)) << 2;
  int s, d;
  if (e < E) { s = ei[e]; d = ei[E + e]; } else { s = d = e - E; }
  float coef = eexp[e] / (ssum[d] + 1e-16f);
  const float4 hv = *(const float4*)(H + ((size_t)s << Dlog2) + c4);
  float* ap = agg + ((size_t)d << Dlog2) + c4;
  atomicAdd(ap + 0, hv.x * coef);
  atomicAdd(ap + 1, hv.y * coef);
  atomicAdd(ap + 2, hv.z * coef);
  atomicAdd(ap + 3, hv.w * coef);
}

__global__ void bias_prelu(float* __restrict__ agg, const float* __restrict__ bias,
                           const float* __restrict__ slope_p, int N, int D) {
  size_t idx = (size_t)blockIdx.x * blockDim.x + threadIdx.x;
  if (idx >= (size_t)N * D) return;
  int d = (int)(idx & (size_t)(D - 1));   // D is a power of two
  float v = agg[idx] + bias[d];
  agg[idx] = (v >= 0.f) ? v : slope_p[0] * v;
}

__global__ void mean_reduce(const float* __restrict__ x, int n, float* __restrict__ out) {
  size_t i = (size_t)blockIdx.x * blockDim.x + threadIdx.x;
  size_t stride = (size_t)gridDim.x * blockDim.x;
  float s = 0.f;
  for (; i < (size_t)n; i += stride) s += x[i];
  for (int o = 16; o; o >>= 1) s += __shfl_down(s, o, 32);
  if ((threadIdx.x & 31) == 0) atomicAdd(out, s * (1.f / (float)n));
}

__global__ void dot_small(const float* __restrict__ a, const float* __restrict__ b,
                          int n, float* __restrict__ out) {
  __shared__ float sh[128];
  int t = threadIdx.x;
  sh[t] = (t < n) ? a[t] * b[t] : 0.f;
  __syncthreads();
  for (int s = 64; s; s >>= 1) { if (t < s) sh[t] += sh[t + s]; __syncthreads(); }
  if (t == 0) *out = sh[0];
}

// softmax fusion over the two branch embeddings; one wave per node
__global__ void fuse(const float* __restrict__ l, const float* __restrict__ g,
                     const float* __restrict__ aw, const float* __restrict__ ab,
                     float* __restrict__ emb, int N) {
  int wid  = (blockIdx.x * blockDim.x + threadIdx.x) >> 5;
  int lane = threadIdx.x & 31;
  if (wid >= N) return;
  const float* lp = l + (size_t)wid * 64;
  const float* gp = g + (size_t)wid * 64;
  float dl = 0.f, dg = 0.f;
  for (int d = lane; d < 64; d += 32) {
    float w = aw[d];
    dl += lp[d] * w;
    dg += gp[d] * w;
  }
  for (int o = 16; o; o >>= 1) { dl += __shfl_down(dl, o, 32); dg += __shfl_down(dg, o, 32); }
  dl = __shfl(dl, 0, 32); dg = __shfl(dg, 0, 32);
  float sl = dl + ab[0], sg = dg + ab[0];
  float mx = fmaxf(sl, sg);
  float el = expf(sl - mx), eg = expf(sg - mx);
  float inv = 1.f / (el + eg);
  float wl = el * inv, wg = eg * inv;
  for (int d = lane; d < 64; d += 32)
    emb[(size_t)wid * 64 + d] = wl * lp[d] + wg * gp[d];
}

__global__ void classifier(const float* __restrict__ emb, const float* __restrict__ cw,
                           const float* __restrict__ cb, float* __restrict__ pred, int N) {
  int i = blockIdx.x * blockDim.x + threadIdx.x;
  if (i >= N) return;
  const float* e = emb + (size_t)i * 64;
  float acc[10];
#pragma unroll
  for (int c = 0; c < 10; ++c) acc[c] = cb[c];
  for (int d = 0; d < 64; ++d) {
    float v = e[d];
#pragma unroll
    for (int c = 0; c < 10; ++c) acc[c] += v * cw[d * 10 + c];
  }
#pragma unroll
  for (int c = 0; c < 10; ++c) pred[(size_t)i * 10 + c] = acc[c];
}

// ---------------------------------------------------------------------------
// host-side orchestration
// ---------------------------------------------------------------------------
static void run_gat_conv(hipStream_t stream, const float* Xin, int K, int D, int Dlog2,
                         const float* W, const float* as, const float* ad,
                         const float* bias, const float* slope,
                         const int* ei, int E, int N,
                         int has_e, const float* ea, const float* mean_p, const float* ce_p,
                         float* Hbuf, float* agg, float* alpha,
                         float* ssrc, float* sdst, float* mmax, float* ssum) {
  int EP = E + N;
  int ntilesN = D / 16;
  int mtiles = (N + 15) / 16;   // N == 50000 is a multiple of 16
  gemm_wmma_f32<<<mtiles, ntilesN * 32, 0, stream>>>(Xin, W, Hbuf, K, D);
  node_scores<<<(N * 32 + 255) / 256, 256, 0, stream>>>(Hbuf, as, ad, ssrc, sdst, N, D);
  init_ms<<<(N + 255) / 256, 256, 0, stream>>>(mmax, ssum, N);
  zero_f32<<<4096, 256, 0, stream>>>(agg, (size_t)N * D);
  edge_alpha_max<<<(EP + 255) / 256, 256, 0, stream>>>(ei, E, N, ssrc, sdst,
                                                       has_e, ea, mean_p, ce_p, alpha, mmax);
  edge_exp<<<(EP + 255) / 256, 256, 0, stream>>>(ei, E, N, alpha, mmax, ssum);
  size_t total4 = ((size_t)EP) << (Dlog2 - 2);
  edge_agg4<<<(unsigned)((total4 + 255) / 256), 256, 0, stream>>>(ei, E, N, Dlog2,
                                                                  Hbuf, alpha, ssum, agg);
  bias_prelu<<<(unsigned)(((size_t)N * D + 255) / 256), 256, 0, stream>>>(agg, bias, slope, N, D);
}

extern "C" void kernel_launch(void* const* d_in, const int* in_sizes, int n_in,
                              void* d_out, int out_size, void* d_ws, size_t ws_size,
                              hipStream_t stream) {
  const float* x    = (const float*)d_in[0];
  const int*   ei   = (const int*)d_in[1];
  const int*   pei  = (const int*)d_in[2];
  const float* pea  = (const float*)d_in[3];
  const float* W1   = (const float*)d_in[4];
  const float* as1  = (const float*)d_in[5];
  const float* ad1  = (const float*)d_in[6];
  const float* b1   = (const float*)d_in[7];
  const float* W2   = (const float*)d_in[8];
  const float* as2  = (const float*)d_in[9];
  const float* ad2  = (const float*)d_in[10];
  const float* b2   = (const float*)d_in[11];
  const float* pra  = (const float*)d_in[12];
  const float* pW1  = (const float*)d_in[13];
  const float* pas1 = (const float*)d_in[14];
  const float* pad1 = (const float*)d_in[15];
  const float* pWe1 = (const float*)d_in[16];
  const float* pae1 = (const float*)d_in[17];
  const float* pb1  = (const float*)d_in[18];
  const float* pW2  = (const float*)d_in[19];
  const float* pas2 = (const float*)d_in[20];
  const float* pad2 = (const float*)d_in[21];
  const float* pWe2 = (const float*)d_in[22];
  const float* pae2 = (const float*)d_in[23];
  const float* pb2  = (const float*)d_in[24];
  const float* ppra = (const float*)d_in[25];
  const float* atw  = (const float*)d_in[26];
  const float* atb  = (const float*)d_in[27];
  const float* cw   = (const float*)d_in[28];
  const float* cb   = (const float*)d_in[29];

  const int DIN = 512, H1 = 128, H2 = 64;
  const int N = in_sizes[0] / DIN;
  const int E = in_sizes[1] / 2;
  const int EP = E + N;

  float* ws = (float*)d_ws;
  size_t off = 0;
  float* H128  = ws + off; off += (size_t)N * H1;   // GEMM output (both layers reuse)
  float* F1    = ws + off; off += (size_t)N * H1;   // f1, later g1
  float* LOUT  = ws + off; off += (size_t)N * H2;
  float* GOUT  = ws + off; off += (size_t)N * H2;
  float* ALPHA = ws + off; off += (size_t)EP;
  float* SSRC  = ws + off; off += N;
  float* SDST  = ws + off; off += N;
  float* MMAX  = ws + off; off += N;
  float* SSUM  = ws + off; off += N;
  float* SCAL  = ws + off; off += 8;   // [0]=mean(ea) [1]=ce1 [2]=ce2

  float* emb  = (float*)d_out;
  float* pred = emb + (size_t)N * H2;

  // scalar prep: mean(ppmi_edge_attr), ce1 = We1·a_e1, ce2 = We2·a_e2
  zero_f32<<<1, 32, 0, stream>>>(SCAL, 8);
  mean_reduce<<<2048, 256, 0, stream>>>(pea, E, &SCAL[0]);
  dot_small<<<1, 128, 0, stream>>>(pWe1, pae1, H1, &SCAL[1]);
  dot_small<<<1, 128, 0, stream>>>(pWe2, pae2, H2, &SCAL[2]);

  // adjacency branch
  run_gat_conv(stream, x,  DIN, H1, 7, W1, as1, ad1, b1, pra, ei, E, N,
               0, pea, &SCAL[0], &SCAL[1], H128, F1, ALPHA, SSRC, SDST, MMAX, SSUM);
  run_gat_conv(stream, F1, H1,  H2, 6, W2, as2, ad2, b2, pra, ei, E, N,
               0, pea, &SCAL[0], &SCAL[1], H128, LOUT, ALPHA, SSRC, SDST, MMAX, SSUM);

  // ppmi branch (edge-attr term collapses to ea[e]*ce)
  run_gat_conv(stream, x,  DIN, H1, 7, pW1, pas1, pad1, pb1, ppra, pei, E, N,
               1, pea, &SCAL[0], &SCAL[1], H128, F1, ALPHA, SSRC, SDST, MMAX, SSUM);
  run_gat_conv(stream, F1, H1,  H2, 6, pW2, pas2, pad2, pb2, ppra, pei, E, N,
               1, pea, &SCAL[0], &SCAL[2], H128, GOUT, ALPHA, SSRC, SDST, MMAX, SSUM);

  // fusion + classifier
  fuse<<<(N * 32 + 255) / 256, 256, 0, stream>>>(LOUT, GOUT, atw, atb, emb, N);
  classifier<<<(N + 255) / 256, 256, 0, stream>>>(emb, cw, cb, pred, N);
}